// SDEBlockLatent_76776835383568
// MI455X (gfx1250) — compile-verified
//
#include <hip/hip_runtime.h>
#include <hip/hip_bf16.h>

typedef __attribute__((ext_vector_type(2))) float v2f;
typedef __attribute__((ext_vector_type(8))) float v8f;

#define BB   2048
#define FSD  32
#define FAD  16
#define TT   512
#define HH   64
#define KIN  48
#define SQRT_DT 0.2236067977499789696f

// Native gfx1250 transcendental tanh if the toolchain exposes it.
#if defined(__has_builtin)
#if __has_builtin(__builtin_amdgcn_tanhf)
#define FAST_TANH(x) __builtin_amdgcn_tanhf(x)
#elif __has_builtin(__builtin_amdgcn_tanh_f32)
#define FAST_TANH(x) __builtin_amdgcn_tanh_f32(x)
#endif
#endif
#ifndef FAST_TANH
#define FAST_TANH(x) tanhf(x)
#endif

// LDS row pitches (in floats). Even (8B-aligned b64 fragment loads) and
// bank-conflict friendly.
#define P_S  50   // s:  16 x 48 (cols 0..15 = a_t, 16..47 = x_t)
#define P_H  66   // W1: 48 x 64, H: 16 x 64
#define P_Y  34   // W2: 64 x 32, Yg: 16 x 32

__global__ __launch_bounds__(256) void sde_scan_kernel(
    const float* __restrict__ in_sig,   // (B, FA, T)
    const float* __restrict__ x0,       // (B, FS)
    const float* __restrict__ noise,    // (T-1, B, FS)
    const float* __restrict__ Wf1, const float* __restrict__ bf1,
    const float* __restrict__ Wf2, const float* __restrict__ bf2,
    const float* __restrict__ Wg1, const float* __restrict__ bg1,
    const float* __restrict__ Wg2, const float* __restrict__ bg2,
    float* __restrict__ out)            // (B, FS, T)
{
    __shared__ __attribute__((aligned(16))) float sW1f[KIN * P_H];
    __shared__ __attribute__((aligned(16))) float sW1g[KIN * P_H];
    __shared__ __attribute__((aligned(16))) float sW2f[HH * P_Y];
    __shared__ __attribute__((aligned(16))) float sW2g[HH * P_Y];
    __shared__ __attribute__((aligned(16))) float sS [16 * P_S];
    __shared__ __attribute__((aligned(16))) float sHf[16 * P_H];
    __shared__ __attribute__((aligned(16))) float sHg[16 * P_H];
    __shared__ __attribute__((aligned(16))) float sYg[16 * P_Y];

    const int tid  = threadIdx.x;
    const int b0   = blockIdx.x * 16;          // 16 batch rows per workgroup
    const int wv   = tid >> 5;                 // wave 0..7
    const int lane = tid & 31;
    const int lo   = lane & 15;                // N / M index within tile
    const int hi   = lane >> 4;                // half-wave selector

    // ---- Preload weights into LDS (loop-invariant) ----
    for (int i = tid; i < KIN * HH; i += 256) {
        int k = i >> 6, n = i & 63;
        sW1f[k * P_H + n] = Wf1[i];
        sW1g[k * P_H + n] = Wg1[i];
    }
    for (int i = tid; i < HH * FSD; i += 256) {
        int k = i >> 5, n = i & 31;
        sW2f[k * P_Y + n] = Wf2[i];
        sW2g[k * P_Y + n] = Wg2[i];
    }
    // ---- x0 into s-state, and out[..., t=0] = x0 ----
    for (int i = tid; i < 16 * FSD; i += 256) {
        int m = i >> 5, n = i & 31;
        float v = x0[(size_t)(b0 + m) * FSD + n];
        sS[m * P_S + 16 + n] = v;
        out[(size_t)(b0 + m) * FSD * TT + (size_t)n * TT] = v;
    }

    // ---- Per-wave roles ----
    const int mlp1 = wv >> 2;                  // layer1: 0 = drift(f), 1 = diff(g)
    const int nt1  = wv & 3;                   // layer1 N-tile (of 4)
    const int mlp2 = wv >> 1;                  // layer2 (waves 0..3 only)
    const int nt2  = wv & 1;                   // layer2 N-tile (of 2)
    const int n1   = nt1 * 16 + lo;            // layer1 output column
    const int n2   = nt2 * 16 + lo;            // layer2 output column
    const float b1v = (mlp1 ? bg1 : bf1)[n1];
    const float b2v = (wv < 4) ? (mlp2 ? bg2 : bf2)[n2] : 0.0f;

    // ---- a_t pipeline: thread -> (row, feature) of the 16x16 action tile ----
    const int am = tid >> 4, aj = tid & 15;
    const size_t abase = ((size_t)(b0 + am) * FAD + aj) * TT;
    float a_cur = in_sig[abase];               // a for step t = 1

    __syncthreads();

    for (int t = 1; t < TT; ++t) {
        // Publish a_{t-1} into s (cols 0..15); prefetch next a.
        sS[am * P_S + aj] = a_cur;
        float a_nxt = (t < TT - 1) ? in_sig[abase + t] : 0.0f;

        // Prefetch eps tile for the combining waves (hidden behind compute).
        float epsr[8];
        if (wv < 2) {
            const size_t ebase = (size_t)(t - 1) * BB * FSD +
                                 (size_t)b0 * FSD + n2;
            #pragma unroll
            for (int v = 0; v < 8; ++v)
                epsr[v] = noise[ebase + (size_t)(v + 8 * hi) * FSD];
        }
        __syncthreads();   // s (a + x) complete

        // ---- Layer 1: h = tanh(s @ W1 + b1), one 16x16 tile per wave ----
        // Two independent accumulation chains (even/odd k-steps) to halve
        // the WMMA RAW-dependency depth on the sequential critical path.
        v8f acc0, acc1;
        #pragma unroll
        for (int v = 0; v < 8; ++v) { acc0[v] = b1v; acc1[v] = 0.0f; }
        const float* W1 = mlp1 ? sW1g : sW1f;
        #pragma unroll
        for (int ks = 0; ks < 6; ++ks) {
            const int kbe = (2 * ks) * 4 + 2 * hi;       // even chain
            const int kbo = (2 * ks + 1) * 4 + 2 * hi;   // odd chain
            v2f ae = *(const v2f*)&sS[lo * P_S + kbe];
            v2f ao = *(const v2f*)&sS[lo * P_S + kbo];
            v2f be, bo;
            be.x = W1[kbe * P_H + n1];  be.y = W1[(kbe + 1) * P_H + n1];
            bo.x = W1[kbo * P_H + n1];  bo.y = W1[(kbo + 1) * P_H + n1];
            acc0 = __builtin_amdgcn_wmma_f32_16x16x4_f32(
                false, ae, false, be, (short)0, acc0, false, false);
            acc1 = __builtin_amdgcn_wmma_f32_16x16x4_f32(
                false, ao, false, bo, (short)0, acc1, false, false);
        }
        {
            float* Hl = mlp1 ? sHg : sHf;
            #pragma unroll
            for (int v = 0; v < 8; ++v)
                Hl[(v + 8 * hi) * P_H + n1] = FAST_TANH(acc0[v] + acc1[v]);
        }
        __syncthreads();   // h complete

        // ---- Layer 2: y = h @ W2 + b2 (waves 0..3, wave-uniform EXEC) ----
        v8f acc2;
        #pragma unroll
        for (int v = 0; v < 8; ++v) acc2[v] = b2v;
        if (wv < 4) {
            v8f acc3;
            #pragma unroll
            for (int v = 0; v < 8; ++v) acc3[v] = 0.0f;
            const float* Hs = mlp2 ? sHg : sHf;
            const float* W2 = mlp2 ? sW2g : sW2f;
            #pragma unroll
            for (int ks = 0; ks < 8; ++ks) {
                const int kbe = (2 * ks) * 4 + 2 * hi;
                const int kbo = (2 * ks + 1) * 4 + 2 * hi;
                v2f ae = *(const v2f*)&Hs[lo * P_H + kbe];
                v2f ao = *(const v2f*)&Hs[lo * P_H + kbo];
                v2f be, bo;
                be.x = W2[kbe * P_Y + n2];  be.y = W2[(kbe + 1) * P_Y + n2];
                bo.x = W2[kbo * P_Y + n2];  bo.y = W2[(kbo + 1) * P_Y + n2];
                acc2 = __builtin_amdgcn_wmma_f32_16x16x4_f32(
                    false, ae, false, be, (short)0, acc2, false, false);
                acc3 = __builtin_amdgcn_wmma_f32_16x16x4_f32(
                    false, ao, false, bo, (short)0, acc3, false, false);
            }
            #pragma unroll
            for (int v = 0; v < 8; ++v) acc2[v] += acc3[v];
            if (mlp2 == 1) {   // diffusion pre-product -> LDS
                #pragma unroll
                for (int v = 0; v < 8; ++v)
                    sYg[(v + 8 * hi) * P_Y + n2] = acc2[v];
            }
        }
        __syncthreads();   // yg complete

        // ---- Combine: x = clip(drift + yg * eps * sqrt(dt)) (waves 0,1) ----
        if (wv < 2) {
            #pragma unroll
            for (int v = 0; v < 8; ++v) {
                const int m = v + 8 * hi;
                float y = acc2[v] + sYg[m * P_Y + n2] * (epsr[v] * SQRT_DT);
                y = fminf(5.0f, fmaxf(-5.0f, y));
                sS[m * P_S + 16 + n2] = y;                        // next state
                out[(size_t)(b0 + m) * FSD * TT + (size_t)n2 * TT + t] = y;
            }
        }
        a_cur = a_nxt;
    }
}

extern "C" void kernel_launch(void* const* d_in, const int* in_sizes, int n_in,
                              void* d_out, int out_size, void* d_ws, size_t ws_size,
                              hipStream_t stream) {
    // setup_inputs order: ts, in_signal, x0, noise, Wf1, bf1, Wf2, bf2, Wg1, bg1, Wg2, bg2
    const float* in_sig = (const float*)d_in[1];
    const float* x0     = (const float*)d_in[2];
    const float* noise  = (const float*)d_in[3];
    const float* Wf1    = (const float*)d_in[4];
    const float* bf1    = (const float*)d_in[5];
    const float* Wf2    = (const float*)d_in[6];
    const float* bf2    = (const float*)d_in[7];
    const float* Wg1    = (const float*)d_in[8];
    const float* bg1    = (const float*)d_in[9];
    const float* Wg2    = (const float*)d_in[10];
    const float* bg2    = (const float*)d_in[11];
    float* out = (float*)d_out;

    dim3 grid(BB / 16);   // 128 workgroups x 16 batch rows
    dim3 block(256);      // 8 waves (wave32)
    sde_scan_kernel<<<grid, block, 0, stream>>>(
        in_sig, x0, noise, Wf1, bf1, Wf2, bf2, Wg1, bg1, Wg2, bg2, out);
}